// AlignerOT_6322191859874
// MI455X (gfx1250) — compile-verified
//
#include <hip/hip_runtime.h>
#include <hip/hip_bf16.h>

typedef float v2f __attribute__((ext_vector_type(2)));
typedef float v8f __attribute__((ext_vector_type(8)));

#define EPSV     0.1f
#define DD       1024
#define NN       128
#define SS       768
#define LOG_A    (-6.9314718055994531f)   /* -ln(1024) */
#define LOG2E    1.4426950408889634f
#define LN2      0.6931471805599453f

// ---------------------------------------------------------------------------
// fp32 WMMA GEMM: one wave computes a 16x32 C strip (two 16x16 tiles sharing
// the A fragment).  C[m,n] = sum_k A[m,k]*B[k,n] (+ bias[n]).
// KC=true : B is K-contiguous per column, element (k,n) = B[n*ldb + k]  (W^T)
// KC=false: B row-major, element (k,n) = B[k*ldb + n]                  (ot)
// V_WMMA_F32_16X16X4_F32 fragment layout (wave32):
//   A 16x4 : lane l<16 -> M=l, K={k0,k0+1}; lane>=16 -> M=l&15, K={k0+2,k0+3}
//   B 4x16 : lane l<16 -> N=l, K={k0,k0+1}; lane>=16 -> N=l&15, K={k0+2,k0+3}
//   C 16x16: VGPR r, lanes 0-15 -> M=r, lanes 16-31 -> M=r+8, N=lane&15
// ---------------------------------------------------------------------------
template <bool KC>
__global__ __launch_bounds__(32)
void wmma_gemm(const float* __restrict__ A, int lda,
               const float* __restrict__ B, int ldb,
               const float* __restrict__ bias,
               float* __restrict__ C, int ldc, int K) {
    const int lane  = threadIdx.x & 31;
    const int half  = lane >> 4;
    const int l     = lane & 15;
    const int kb    = half * 2;            // K sub-offset inside the fragment
    const int tileM = blockIdx.y * 16;
    const int tileN = blockIdx.x * 32;

    v8f acc0 = {}, acc1 = {};
    const float* Arow = A + (size_t)(tileM + l) * lda + kb;

    #pragma unroll 4
    for (int k0 = 0; k0 < K; k0 += 4) {
        __builtin_prefetch(Arow + k0 + 64, 0, 0);   // global_prefetch_b8
        v2f a = *(const v2f*)(Arow + k0);
        v2f b0, b1;
        if (KC) {
            b0 = *(const v2f*)(B + (size_t)(tileN + l) * ldb + k0 + kb);
            b1 = *(const v2f*)(B + (size_t)(tileN + 16 + l) * ldb + k0 + kb);
        } else {
            const float* Bp = B + (size_t)(k0 + kb) * ldb;
            b0.x = Bp[tileN + l];
            b0.y = Bp[(size_t)ldb + tileN + l];
            b1.x = Bp[tileN + 16 + l];
            b1.y = Bp[(size_t)ldb + tileN + 16 + l];
        }
        acc0 = __builtin_amdgcn_wmma_f32_16x16x4_f32(false, a, false, b0,
                                                     (short)0, acc0, false, false);
        acc1 = __builtin_amdgcn_wmma_f32_16x16x4_f32(false, a, false, b1,
                                                     (short)0, acc1, false, false);
    }

    const float bv0 = bias ? bias[tileN + l] : 0.0f;
    const float bv1 = bias ? bias[tileN + 16 + l] : 0.0f;
    #pragma unroll
    for (int r = 0; r < 8; ++r) {
        const int row = tileM + r + half * 8;
        C[(size_t)row * ldc + tileN + l]      = acc0[r] + bv0;
        C[(size_t)row * ldc + tileN + 16 + l] = acc1[r] + bv1;
    }
}

// ---------------------------------------------------------------------------
// Per-sample log-domain Sinkhorn, 50 iterations. One 1024-thread block
// (32 waves) per sample; one row/column per thread per phase.
// Inner loop: ds_load_b64 (packed {A2,B2}) + v_fma + v_max  (pass 1)
//             ds_load_b64 + v_fma + v_exp + v_add           (pass 2)
// using (g[k]-300(s_k-y)^2)/eps * log2e = A2[k] + B2[k]*y + c2(y).
// ---------------------------------------------------------------------------
__global__ __launch_bounds__(1024)
void sinkhorn_kernel(const float* __restrict__ srcg, const float* __restrict__ Yg,
                     float* __restrict__ Fg, float* __restrict__ Gg) {
    __shared__ float  sS[DD], sY[DD], sF[DD], sG[DD];
    __shared__ float2 sAB[DD];
    const int i   = blockIdx.x;
    const int tid = threadIdx.x;

    sS[tid] = srcg[(size_t)i * DD + tid];
    sY[tid] = Yg[(size_t)i * DD + tid];
    sG[tid] = 0.0f;
    __syncthreads();

    for (int it = 0; it < 50; ++it) {
        // ---- f update ----
        {
            const float s = sS[tid];
            sAB[tid] = make_float2((10.0f * sG[tid] - 3000.0f * s * s) * LOG2E,
                                   (6000.0f * LOG2E) * s);
        }
        __syncthreads();
        {
            const float y = sY[tid];
            float m = -3.4e38f;
            #pragma unroll 8
            for (int k = 0; k < DD; ++k) {
                const float2 ab = sAB[k];
                m = fmaxf(m, fmaf(ab.y, y, ab.x));
            }
            float ssum = 0.0f;
            #pragma unroll 8
            for (int k = 0; k < DD; ++k) {
                const float2 ab = sAB[k];
                ssum += exp2f(fmaf(ab.y, y, ab.x) - m);
            }
            const float lse = LN2 * ((m - (3000.0f * LOG2E) * y * y) + __log2f(ssum));
            sF[tid] = EPSV * (LOG_A - lse);
        }
        __syncthreads();
        // ---- g update ----
        {
            const float y = sY[tid];
            sAB[tid] = make_float2((10.0f * sF[tid] - 3000.0f * y * y) * LOG2E,
                                   (6000.0f * LOG2E) * y);
        }
        __syncthreads();
        {
            const float x = sS[tid];
            float m = -3.4e38f;
            #pragma unroll 8
            for (int j = 0; j < DD; ++j) {
                const float2 ab = sAB[j];
                m = fmaxf(m, fmaf(ab.y, x, ab.x));
            }
            float ssum = 0.0f;
            #pragma unroll 8
            for (int j = 0; j < DD; ++j) {
                const float2 ab = sAB[j];
                ssum += exp2f(fmaf(ab.y, x, ab.x) - m);
            }
            const float lse = LN2 * ((m - (3000.0f * LOG2E) * x * x) + __log2f(ssum));
            sG[tid] = EPSV * (LOG_A - lse);
        }
        __syncthreads();
    }

    Fg[(size_t)i * DD + tid] = sF[tid];
    Gg[(size_t)i * DD + tid] = sG[tid];
}

// ---------------------------------------------------------------------------
// ot[j,k] = (1/N)*sum_i exp((f_i[j]+g_i[k]-cost_i[j,k])/eps) * D*SCALE + delta
// 16x16 output tile per block; 16 samples staged through LDS per chunk
// (coalesced loads, 2 barriers per 16 samples).
// ---------------------------------------------------------------------------
__global__ __launch_bounds__(256)
void ot_kernel(const float* __restrict__ srcg, const float* __restrict__ Yg,
               const float* __restrict__ Fg, const float* __restrict__ Gg,
               const float* __restrict__ delta, float* __restrict__ ot) {
    __shared__ float tS[16][16], tG[16][16], tY[16][16], tF[16][16];
    const int tx = threadIdx.x & 15;      // k (src/col)
    const int ty = threadIdx.x >> 4;      // j (Y/row)
    const int k0 = blockIdx.x * 16;
    const int j0 = blockIdx.y * 16;

    float sum = 0.0f;
    for (int i0 = 0; i0 < NN; i0 += 16) {
        __syncthreads();
        tS[ty][tx] = srcg[(size_t)(i0 + ty) * DD + k0 + tx];
        tG[ty][tx] = Gg[(size_t)(i0 + ty) * DD + k0 + tx];
        tY[ty][tx] = Yg[(size_t)(i0 + ty) * DD + j0 + tx];
        tF[ty][tx] = Fg[(size_t)(i0 + ty) * DD + j0 + tx];
        __syncthreads();
        #pragma unroll
        for (int ii = 0; ii < 16; ++ii) {
            const float d = tS[ii][tx] - tY[ii][ty];
            const float v2 = (10.0f * LOG2E) * (tF[ii][ty] + tG[ii][tx])
                           - (3000.0f * LOG2E) * d * d;
            sum += exp2f(v2);
        }
    }
    const size_t idx = (size_t)(j0 + ty) * DD + (k0 + tx);
    ot[idx] = sum * 2400.0f + delta[idx];   // mean over N * D * SCALE = 2400
}

// ---------------------------------------------------------------------------
extern "C" void kernel_launch(void* const* d_in, const int* in_sizes, int n_in,
                              void* d_out, int out_size, void* d_ws, size_t ws_size,
                              hipStream_t stream) {
    const float* X     = (const float*)d_in[0];   // [128, 768]
    const float* Y     = (const float*)d_in[1];   // [128, 1024]
    const float* W     = (const float*)d_in[2];   // [1024, 768]
    const float* bias  = (const float*)d_in[3];   // [1024]
    const float* delta = (const float*)d_in[4];   // [1024, 1024]
    float* out = (float*)d_out;                   // [128, 1024]

    float* ws  = (float*)d_ws;
    float* src = ws;                        // 128*1024
    float* F   = src + (size_t)NN * DD;     // 128*1024
    float* G   = F   + (size_t)NN * DD;     // 128*1024
    float* ot  = G   + (size_t)NN * DD;     // 1024*1024

    dim3 gemmGrid(DD / 32, NN / 16);
    // 1) src = X @ W^T + b    (B(k,n) = W[n*S + k], K-contiguous)
    wmma_gemm<true><<<gemmGrid, 32, 0, stream>>>(X, SS, W, SS, bias, src, DD, SS);

    // 2) per-sample Sinkhorn (50 iters)
    sinkhorn_kernel<<<NN, 1024, 0, stream>>>(src, Y, F, G);

    // 3) ot_matrix = mean_i P_i * D * SCALE + delta_ot
    ot_kernel<<<dim3(DD / 16, DD / 16), 256, 0, stream>>>(src, Y, F, G, delta, ot);

    // 4) out = src @ ot_matrix  (B row-major)
    wmma_gemm<false><<<gemmGrid, 32, 0, stream>>>(src, DD, ot, DD, nullptr, out, DD, DD);
}